// VariationalGraphDecoder_34497177322135
// MI455X (gfx1250) — compile-verified
//
#include <hip/hip_runtime.h>
#include <math.h>

#define N_NODES 50000
#define N_EDGES 800000
#define HIDDEN  128
#define OUT_DIM 64
#define MTILES  (N_NODES / 16)   // 3125 exactly
#define LDA     (HIDDEN + 4)     // LDS row stride in floats: 132 -> bank-conflict-free

typedef __attribute__((ext_vector_type(2))) float v2f;
typedef __attribute__((ext_vector_type(8))) float v8f;

// ---------------------------------------------------------------------------
// f32 WMMA GEMM with register-resident B panel and persistent blocks.
// C[M x N] = act(A[M x 128] @ W[128 x N] (+ bias))
// One wave owns one 16-wide N tile; its full B panel (16 cols x 128 K) lives
// in 32 v2f registers for the whole kernel. Blocks grid-stride over M tiles;
// per tile only the 16x128 A slab moves (float4 -> LDS, padded stride).
// ---------------------------------------------------------------------------
template <bool BIAS, bool RELU, int N>
__global__ __launch_bounds__((N / 16) * 32) void gemm_f32_wmma(
    const float* __restrict__ A,
    const float* __restrict__ W,
    const float* __restrict__ bias,
    float* __restrict__ C)
{
    __shared__ float As[16 * LDA];

    const int tid  = threadIdx.x;
    const int wave = tid >> 5;
    const int lane = tid & 31;
    const int half = lane >> 4;   // 0: lanes 0-15, 1: lanes 16-31
    const int l15  = lane & 15;
    const int n0   = wave * 16;
    const int col  = n0 + l15;

    // --- hoist B panel into registers: 32 k-steps x v2f = 64 VGPRs ---------
    v2f Breg[32];
#pragma unroll
    for (int t = 0; t < 32; ++t) {
        const int k = 4 * t + 2 * half;
        Breg[t].x = W[(size_t)k * N + col];
        Breg[t].y = W[(size_t)(k + 1) * N + col];
    }
    const float bv = BIAS ? bias[col] : 0.0f;

    const int nthreads = (N / 16) * 32;

    // --- persistent loop over M tiles --------------------------------------
    for (int mt = blockIdx.x; mt < MTILES; mt += gridDim.x) {
        const int m0 = mt * 16;
        __syncthreads();   // As readers from previous tile are done
        // stage 16x128 A slab as float4 (512 vec4 elements)
        for (int i = tid; i < 16 * 32; i += nthreads) {
            const int r  = i >> 5;       // row 0..15
            const int c4 = i & 31;       // float4 column
            const float4 v = ((const float4*)(A + (size_t)(m0 + r) * HIDDEN))[c4];
            ((float4*)&As[r * LDA])[c4] = v;
        }
        __syncthreads();

        v8f acc = {};
#pragma unroll
        for (int t = 0; t < 32; ++t) {
            const int k = 4 * t + 2 * half;
            v2f a;
            a.x = As[l15 * LDA + k];
            a.y = As[l15 * LDA + k + 1];
            acc = __builtin_amdgcn_wmma_f32_16x16x4_f32(false, a, false, Breg[t],
                                                        (short)0, acc, false, false);
        }

        // epilogue: VGPR v holds (row = m0 + v + 8*half, col)
#pragma unroll
        for (int v = 0; v < 8; ++v) {
            const int row = m0 + v + 8 * half;
            float val = acc[v] + bv;
            if (RELU) val = fmaxf(val, 0.0f);
            C[(size_t)row * N + col] = val;
        }
    }
}

// ---------------------------------------------------------------------------
// Output GEMM: out = relu(AGG + bg) @ W2 + b2, sigmoid on global column 0.
// Same structure, N=64 (4 waves). relu+bg fused into the float4 LDS staging.
// ---------------------------------------------------------------------------
__global__ __launch_bounds__(128) void gemm_out_kernel(const float* __restrict__ AGG,
                                                       const float* __restrict__ bg,
                                                       const float* __restrict__ W2,
                                                       const float* __restrict__ b2,
                                                       float* __restrict__ out)
{
    __shared__ float As[16 * LDA];

    const int tid  = threadIdx.x;
    const int wave = tid >> 5;
    const int lane = tid & 31;
    const int half = lane >> 4;
    const int l15  = lane & 15;
    const int n0   = wave * 16;
    const int col  = n0 + l15;

    v2f Breg[32];
#pragma unroll
    for (int t = 0; t < 32; ++t) {
        const int k = 4 * t + 2 * half;
        Breg[t].x = W2[(size_t)k * OUT_DIM + col];
        Breg[t].y = W2[(size_t)(k + 1) * OUT_DIM + col];
    }
    const float bv = b2[col];

    for (int mt = blockIdx.x; mt < MTILES; mt += gridDim.x) {
        const int m0 = mt * 16;
        __syncthreads();
        for (int i = tid; i < 16 * 32; i += 128) {
            const int r  = i >> 5;
            const int c4 = i & 31;
            float4 v = ((const float4*)(AGG + (size_t)(m0 + r) * HIDDEN))[c4];
            const float4 g = ((const float4*)bg)[c4];
            v.x = fmaxf(v.x + g.x, 0.0f);
            v.y = fmaxf(v.y + g.y, 0.0f);
            v.z = fmaxf(v.z + g.z, 0.0f);
            v.w = fmaxf(v.w + g.w, 0.0f);
            ((float4*)&As[r * LDA])[c4] = v;
        }
        __syncthreads();

        v8f acc = {};
#pragma unroll
        for (int t = 0; t < 32; ++t) {
            const int k = 4 * t + 2 * half;
            v2f a;
            a.x = As[l15 * LDA + k];
            a.y = As[l15 * LDA + k + 1];
            acc = __builtin_amdgcn_wmma_f32_16x16x4_f32(false, a, false, Breg[t],
                                                        (short)0, acc, false, false);
        }

#pragma unroll
        for (int v = 0; v < 8; ++v) {
            const int row = m0 + v + 8 * half;
            float val = acc[v] + bv;
            if (col == 0) val = 1.0f / (1.0f + __expf(-val));  // sigmoid on out[:,0]
            out[(size_t)row * OUT_DIM + col] = val;
        }
    }
}

// ---------------------------------------------------------------------------
// Edge / pointwise kernels
// ---------------------------------------------------------------------------
__global__ void zero_f32(float* __restrict__ p, int n)
{
    const int i = blockIdx.x * blockDim.x + threadIdx.x;
    if (i < n) p[i] = 0.0f;
}

__global__ void deg_kernel(const long long* __restrict__ ei, float* __restrict__ deg)
{
    const int e = blockIdx.x * blockDim.x + threadIdx.x;
    if (e < N_EDGES) {
        const int c = (int)ei[(size_t)N_EDGES + e];   // target
        atomicAdd(&deg[c], 1.0f);
    }
}

__global__ void dinv_kernel(float* __restrict__ deg)
{
    const int i = blockIdx.x * blockDim.x + threadIdx.x;
    if (i < N_NODES) deg[i] = rsqrtf(deg[i] + 1.0f);  // +1 = self loop; always > 0
}

// AGG[i][:] = XW[i][:] * dinv[i]^2   (self-loop term, also zero-initializes AGG)
__global__ void init_agg_kernel(const float* __restrict__ XW,
                                const float* __restrict__ dinv,
                                float* __restrict__ AGG)
{
    const int i = blockIdx.x * blockDim.x + threadIdx.x;   // float4 granularity
    if (i < N_NODES * (HIDDEN / 4)) {
        const int node = i / (HIDDEN / 4);
        float s = dinv[node];
        s = s * s;
        const float4 v = ((const float4*)XW)[i];
        float4 r;
        r.x = v.x * s; r.y = v.y * s; r.z = v.z * s; r.w = v.w * s;
        ((float4*)AGG)[i] = r;
    }
}

// one wave per edge; each lane scatters a float4 chunk of the 128-wide row
__global__ __launch_bounds__(256) void scatter_kernel(const long long* __restrict__ ei,
                                                      const float* __restrict__ XW,
                                                      const float* __restrict__ dinv,
                                                      float* __restrict__ AGG)
{
    const int gid  = blockIdx.x * blockDim.x + threadIdx.x;
    const int e    = gid >> 5;
    const int lane = gid & 31;
    if (e < N_EDGES) {
        const int r = (int)ei[e];                       // source j
        const int c = (int)ei[(size_t)N_EDGES + e];     // target i
        const float norm = dinv[r] * dinv[c];
        const float4 v = ((const float4*)(XW + (size_t)r * HIDDEN))[lane];
        float* dst = AGG + (size_t)c * HIDDEN + lane * 4;
        atomicAdd(dst + 0, v.x * norm);
        atomicAdd(dst + 1, v.y * norm);
        atomicAdd(dst + 2, v.z * norm);
        atomicAdd(dst + 3, v.w * norm);
    }
}

// ---------------------------------------------------------------------------
// Launch
// inputs: z, W1, b1, Wg, bg, W2, b2, edge_index(int64 [2, N_EDGES])
// ---------------------------------------------------------------------------
extern "C" void kernel_launch(void* const* d_in, const int* in_sizes, int n_in,
                              void* d_out, int out_size, void* d_ws, size_t ws_size,
                              hipStream_t stream)
{
    const float*     z  = (const float*)d_in[0];
    const float*     W1 = (const float*)d_in[1];
    const float*     b1 = (const float*)d_in[2];
    const float*     Wg = (const float*)d_in[3];
    const float*     bg = (const float*)d_in[4];
    const float*     W2 = (const float*)d_in[5];
    const float*     b2 = (const float*)d_in[6];
    const long long* ei = (const long long*)d_in[7];
    float*           out = (float*)d_out;

    char* ws = (char*)d_ws;
    const size_t featBytes = (size_t)N_NODES * HIDDEN * sizeof(float);  // 25.6 MB
    float* H1  = (float*)ws;                      // later reused as AGG
    float* XW  = (float*)(ws + featBytes);
    float* deg = (float*)(ws + 2 * featBytes);    // becomes dinv in place

    const int gemm_blocks = 1024;   // persistent blocks, grid-stride over 3125 M tiles

    // h1 = relu(z @ W1 + b1)
    hipLaunchKernelGGL((gemm_f32_wmma<true, true, HIDDEN>), dim3(gemm_blocks), dim3(256),
                       0, stream, z, W1, b1, H1);

    // degree -> dinv
    zero_f32<<<(N_NODES + 255) / 256, 256, 0, stream>>>(deg, N_NODES);
    deg_kernel<<<(N_EDGES + 255) / 256, 256, 0, stream>>>(ei, deg);
    dinv_kernel<<<(N_NODES + 255) / 256, 256, 0, stream>>>(deg);

    // xw = h1 @ Wg
    hipLaunchKernelGGL((gemm_f32_wmma<false, false, HIDDEN>), dim3(gemm_blocks), dim3(256),
                       0, stream, H1, Wg, (const float*)nullptr, XW);

    // H1 buffer is now dead -> reuse as AGG
    float* AGG = H1;
    init_agg_kernel<<<(N_NODES * (HIDDEN / 4) + 255) / 256, 256, 0, stream>>>(XW, deg, AGG);

    // scatter-add over edges (32 lanes per edge)
    const long long scatter_threads = (long long)N_EDGES * 32;
    scatter_kernel<<<(unsigned)((scatter_threads + 255) / 256), 256, 0, stream>>>(ei, XW, deg, AGG);

    // out = relu(AGG + bg) @ W2 + b2; sigmoid(out[:,0])
    gemm_out_kernel<<<gemm_blocks, 128, 0, stream>>>(AGG, bg, W2, b2, out);
}